// TensorDLT_40527311405293
// MI455X (gfx1250) — compile-verified
//
#include <hip/hip_runtime.h>
#include <cstdint>
#include <cstddef>

#define TPB 256

typedef __attribute__((ext_vector_type(4))) unsigned int u32x4;
typedef __attribute__((ext_vector_type(4))) int          i32x4;
typedef __attribute__((ext_vector_type(8))) int          i32x8;

#ifndef __has_builtin
#define __has_builtin(x) 0
#endif

#if __has_builtin(__builtin_amdgcn_tensor_load_to_lds) && \
    __has_builtin(__builtin_amdgcn_s_wait_tensorcnt)
#define USE_TDM 1
#else
#define USE_TDM 0
#endif

// One thread per batch item. Input tile (TPB*8 f32 = 8KB) staged through LDS
// by the Tensor Data Mover (one descriptor per workgroup, issued by wave 0,
// completion via s_wait_tensorcnt + workgroup barrier). The 8x8 DLT system is
// solved analytically via the constant-block reduction (see launch comment),
// keeping everything in registers — no pivot search, no scratch.
__global__ __launch_bounds__(TPB) void tensordlt_kernel(
    const float* __restrict__ Hx,       // (B, 8) perturbations
    const float* __restrict__ corners,  // (8,) batch-uniform corners
    float* __restrict__ out,            // (B, 8) solution h
    int nbatch)
{
    const int gid = blockIdx.x * TPB + threadIdx.x;

#if USE_TDM
    __shared__ __align__(16) float tile[TPB * 8];

    if ((threadIdx.x >> 5) == 0) {  // wave 0 only (uniform branch)
        const unsigned long long gaddr =
            (unsigned long long)(uintptr_t)Hx +
            (unsigned long long)blockIdx.x * (TPB * 8ull * 4ull);
        const unsigned int ldsa = (unsigned int)(uintptr_t)tile;  // LDS byte offset
        const unsigned int td0  = (unsigned int)nbatch * 8u;      // tensor_dim0 (elems)

        // ---- D# group 0 (128b): count=1, lds_addr, global_addr, type=2 ----
        u32x4 g0;
        g0[0] = 1u;                                              // count=1, user mode
        g0[1] = ldsa;                                            // lds_addr [63:32]
        g0[2] = (unsigned int)(gaddr & 0xFFFFFFFFull);           // global_addr lo
        g0[3] = (unsigned int)((gaddr >> 32) & 0x1FFFFFFull)     // global_addr[56:32]
              | (2u << 30);                                      // type = 2 ("image")

        // ---- D# group 1 (256b) ----
        i32x8 g1;
        g1[0] = (int)(2u << 16);                       // data_size = 2 -> 4 bytes
        g1[1] = (int)((td0 & 0xFFFFu) << 16);          // tensor_dim0[15:0] @ [63:48]
        g1[2] = (int)((td0 >> 16) | (1u << 16));       // tensor_dim0[31:16], tensor_dim1=1
        g1[3] = (int)((unsigned)(TPB * 8) << 16);      // tile_dim0 = 2048 @ [127:112]
        g1[4] = 1;                                     // tile_dim1 = 1, tile_dim2 = 0
        g1[5] = (int)(unsigned)(TPB * 8);              // tensor_dim0_stride lo
        g1[6] = 0;                                     // stride0 hi, stride1 lo
        g1[7] = 0;                                     // stride1 hi
        i32x4 g2 = {0, 0, 0, 0};                       // <=2D tensor: groups 2/3 unused
        i32x4 g3 = {0, 0, 0, 0};
#if __clang_major__ >= 23
        i32x8 g4 = {0, 0, 0, 0, 0, 0, 0, 0};
        __builtin_amdgcn_tensor_load_to_lds(g0, g1, g2, g3, g4, 0);
#else
        __builtin_amdgcn_tensor_load_to_lds(g0, g1, g2, g3, 0);
#endif
        __builtin_amdgcn_s_wait_tensorcnt((short)0);   // TDM done -> data in LDS
    }
    __syncthreads();

    const float4 hv0 = *reinterpret_cast<const float4*>(&tile[threadIdx.x * 8 + 0]);
    const float4 hv1 = *reinterpret_cast<const float4*>(&tile[threadIdx.x * 8 + 4]);
#else
    float4 hv0 = {}, hv1 = {};
    if (gid < nbatch) {
        hv0 = *reinterpret_cast<const float4*>(Hx + (size_t)gid * 8 + 0);
        hv1 = *reinterpret_cast<const float4*>(Hx + (size_t)gid * 8 + 4);
    }
#endif

    if (gid >= nbatch) return;

    // Batch-uniform corner coordinates (scalar loads).
    const float x0 = corners[0], y0 = corners[1];
    const float x1 = corners[2], y1 = corners[3];
    const float x2 = corners[4], y2 = corners[5];
    const float x3 = corners[6], y3 = corners[7];

    // Cb = corners + H ; u = Cb[0::2], v = Cb[1::2]
    const float u0 = x0 + hv0.x, v0 = y0 + hv0.y;
    const float u1 = x1 + hv0.z, v1 = y1 + hv0.w;
    const float u2 = x2 + hv1.x, v2 = y2 + hv1.y;
    const float u3 = x3 + hv1.z, v3 = y3 + hv1.w;

    // w: 4-vector orthogonal to the constant columns (x_i), (y_i), (1).
    // w_i = +/- 3x3 minor (generalized cross product).
    auto det3 = [](float xa, float ya, float xb, float yb, float xc, float yc) {
        return (xb - xa) * (yc - ya) - (xc - xa) * (yb - ya);
    };
    const float w0 =  det3(x1, y1, x2, y2, x3, y3);
    const float w1 = -det3(x0, y0, x2, y2, x3, y3);
    const float w2 =  det3(x0, y0, x1, y1, x3, y3);
    const float w3 = -det3(x0, y0, x1, y1, x2, y2);

    const float a0 = w0 * x0, a1 = w1 * x1, a2 = w2 * x2, a3 = w3 * x3;
    const float b0 = w0 * y0, b1 = w1 * y1, b2 = w2 * y2, b3 = w3 * y3;

    // Reduced 2x2 system in (h7, h8):
    //  even rows:  p h7 + q h8 = -r
    //  odd  rows:  s h7 + t h8 =  m
    const float p = a0 * v0 + a1 * v1 + a2 * v2 + a3 * v3;
    const float q = b0 * v0 + b1 * v1 + b2 * v2 + b3 * v3;
    const float r = w0 * v0 + w1 * v1 + w2 * v2 + w3 * v3;
    const float s = a0 * u0 + a1 * u1 + a2 * u2 + a3 * u3;
    const float t = b0 * u0 + b1 * u1 + b2 * u2 + b3 * u3;
    const float m = w0 * u0 + w1 * u1 + w2 * u2 + w3 * u3;

    const float inv2 = 1.0f / (p * t - q * s);
    const float h7 = (-r * t - q * m) * inv2;
    const float h8 = ( p * m + r * s) * inv2;

    // Back-substitution: M [h1 h2 h3]^T = g, M [h4 h5 h6]^T = f,
    // with constant M = [[x0,y0,1],[x1,y1,1],[x2,y2,1]] (closed-form inverse).
    const float d0 = x0 * h7 + y0 * h8;
    const float d1 = x1 * h7 + y1 * h8;
    const float d2 = x2 * h7 + y2 * h8;
    const float g0r = u0 * (d0 - 1.0f);
    const float g1r = u1 * (d1 - 1.0f);
    const float g2r = u2 * (d2 - 1.0f);
    const float f0r = v0 * (d0 + 1.0f);
    const float f1r = v1 * (d1 + 1.0f);
    const float f2r = v2 * (d2 + 1.0f);

    const float C00 = y1 - y2,            C10 = y2 - y0,            C20 = y0 - y1;
    const float C01 = x2 - x1,            C11 = x0 - x2,            C21 = x1 - x0;
    const float C02 = x1 * y2 - x2 * y1,  C12 = x2 * y0 - x0 * y2,  C22 = x0 * y1 - x1 * y0;
    const float invM = 1.0f / (x0 * C00 + y0 * C01 + C02);

    const float h1 = (C00 * g0r + C10 * g1r + C20 * g2r) * invM;
    const float h2 = (C01 * g0r + C11 * g1r + C21 * g2r) * invM;
    const float h3 = (C02 * g0r + C12 * g1r + C22 * g2r) * invM;
    const float h4 = (C00 * f0r + C10 * f1r + C20 * f2r) * invM;
    const float h5 = (C01 * f0r + C11 * f1r + C21 * f2r) * invM;
    const float h6 = (C02 * f0r + C12 * f1r + C22 * f2r) * invM;

    float4* op = reinterpret_cast<float4*>(out + (size_t)gid * 8);
    op[0] = make_float4(h1, h2, h3, h4);
    op[1] = make_float4(h5, h6, h7, h8);
}

extern "C" void kernel_launch(void* const* d_in, const int* in_sizes, int n_in,
                              void* d_out, int out_size, void* d_ws, size_t ws_size,
                              hipStream_t stream) {
    const float* Hx      = (const float*)d_in[0];   // (B, 8) float32
    const float* corners = (const float*)d_in[1];   // (8,)   float32
    float* out = (float*)d_out;                     // (B, 8) float32

    const int nbatch = in_sizes[0] / 8;
    const int grid = (nbatch + TPB - 1) / TPB;
    tensordlt_kernel<<<grid, TPB, 0, stream>>>(Hx, corners, out, nbatch);
}